// VAR_Layer_31748398252614
// MI455X (gfx1250) — compile-verified
//
#include <hip/hip_runtime.h>
#include <hip/hip_bf16.h>
#include <math.h>

typedef float v2f __attribute__((ext_vector_type(2)));
typedef float v8f __attribute__((ext_vector_type(8)));

#define D_DIM 8
#define T_LEN 128
#define N_DT 1024           // D*T
#define KM 16384            // K_BATCH * M_BATCH
#define P_LAGS 2

// ---------------------------------------------------------------------------
// Kernel 1: gamma recursion + row0 recursion (serial in t/s, 64 threads).
// Writes gamma[T][8][8] to global and Z row-0 / col-0 blocks into Zcov.
// ---------------------------------------------------------------------------
__global__ __launch_bounds__(64) void init_cov(const float* __restrict__ A,
                                               const float* __restrict__ log_sigma,
                                               float* __restrict__ Zcov,
                                               float* __restrict__ gamma)
{
    __shared__ float As[P_LAGS * 64];
    __shared__ float sig[D_DIM];
    __shared__ float gr[2][64];   // rolling gamma[t-1], gamma[t-2]
    __shared__ float rr[2][64];   // rolling row0[s-1], row0[s-2]

    const int tid = threadIdx.x;
    const int i = tid >> 3, j = tid & 7;

    for (int x = tid; x < P_LAGS * 64; x += 64) As[x] = A[x];
    if (tid < D_DIM) sig[tid] = expf(log_sigma[tid]);
    __syncthreads();

    // ---- gamma recursion: gamma[0]=diag(sigma); gamma[t]=sum_k A[k-1]@gamma[t-k]
    float g0 = (i == j) ? sig[i] : 0.0f;
    gr[0][tid] = g0;
    gamma[tid] = g0;
    __syncthreads();
    for (int t = 1; t < T_LEN; ++t) {
        float acc = 0.0f;
        const int kmax = (t < P_LAGS) ? t : P_LAGS;
        for (int k = 1; k <= kmax; ++k) {
            const float* gp = gr[(t - k) & 1];
            for (int m = 0; m < 8; ++m)
                acc += As[(k - 1) * 64 + i * 8 + m] * gp[m * 8 + j];
        }
        __syncthreads();
        gr[t & 1][tid] = acc;
        gamma[t * 64 + tid] = acc;
        __syncthreads();
    }

    // ---- row0 recursion: row0[0]=I; row0[s]=sum_k row0[s-k] @ A[k-1]^T
    float r0 = (i == j) ? 1.0f : 0.0f;
    rr[0][tid] = r0;
    Zcov[(size_t)i * N_DT + j] = r0;            // Z[0][0]
    __syncthreads();
    for (int s = 1; s < T_LEN; ++s) {
        float acc = 0.0f;
        const int kmax = (s < P_LAGS) ? s : P_LAGS;
        for (int k = 1; k <= kmax; ++k) {
            const float* rp = rr[(s - k) & 1];
            for (int m = 0; m < 8; ++m)
                acc += rp[i * 8 + m] * As[(k - 1) * 64 + j * 8 + m];
        }
        __syncthreads();
        rr[s & 1][tid] = acc;
        Zcov[(size_t)i * N_DT + s * 8 + j] = acc;          // Z[0][s][i][j]
        Zcov[(size_t)(s * 8 + j) * N_DT + i] = acc;        // Z[s][0] = row0[s]^T
        __syncthreads();
    }
}

// ---------------------------------------------------------------------------
// Kernel 2 (launched T-1 times, stream-ordered): Z[t][s] for s = t..T-1.
// Reads rows t-1, t-2 (complete from previous launches) + gamma.
// ---------------------------------------------------------------------------
__global__ __launch_bounds__(64) void cov_row(int t,
                                              const float* __restrict__ A,
                                              const float* __restrict__ log_sigma,
                                              float* __restrict__ Zcov,
                                              const float* __restrict__ gamma)
{
    __shared__ float As[P_LAGS * 64];
    const int s = t + blockIdx.x;
    const int tid = threadIdx.x;
    const int i = tid >> 3, j = tid & 7;

    for (int x = tid; x < P_LAGS * 64; x += 64) As[x] = A[x];
    __syncthreads();

    float acc = 0.0f;
    const int ktmax = (t < P_LAGS) ? t : P_LAGS;
    for (int kt = 1; kt <= ktmax; ++kt) {
        for (int ks = 1; ks <= P_LAGS; ++ks) {
            const int ss = s - ks;
            if (ss < 0) continue;
            const float* g = Zcov + (size_t)((t - kt) * 8) * N_DT + ss * 8;
            for (int a = 0; a < 8; ++a) {
                float inner = 0.0f;
                for (int b = 0; b < 8; ++b)
                    inner += g[a * N_DT + b] * As[(ks - 1) * 64 + j * 8 + b];
                acc += As[(kt - 1) * 64 + i * 8 + a] * inner;
            }
        }
    }
    if (s == t && i == j) acc += expf(log_sigma[i]);       // Sigma_eps on diagonal
    for (int k = 1; k <= P_LAGS; ++k) {
        const int ss = s - k;
        if (ss < t) continue;
        const float* g = gamma + (ss - t) * 64;            // gamma^T @ A^T
        float inner = 0.0f;
        for (int a = 0; a < 8; ++a)
            inner += g[a * 8 + i] * As[(k - 1) * 64 + j * 8 + a];
        acc += inner;
    }

    Zcov[(size_t)(t * 8 + i) * N_DT + s * 8 + j] = acc;
    if (s > t)
        Zcov[(size_t)(s * 8 + j) * N_DT + t * 8 + i] = acc;   // mirror transpose
}

// ---------------------------------------------------------------------------
// Kernel 3: in-place Cholesky of Zcov (1024x1024), one 1024-thread workgroup.
// Accumulates ldj = sum log|diag(L)|, zeroes the upper triangle for the GEMM.
// ---------------------------------------------------------------------------
__global__ __launch_bounds__(1024) void cholesky(float* __restrict__ Zc,
                                                 float* __restrict__ ldj_out)
{
    const int N = N_DT;
    const int tid = threadIdx.x;
    __shared__ float sinv;
    float lacc = 0.0f;

    for (int k = 0; k < N; ++k) {
        if (tid == 0) {
            float dv = sqrtf(Zc[(size_t)k * N + k]);
            Zc[(size_t)k * N + k] = dv;
            lacc += logf(fabsf(dv));
            sinv = 1.0f / dv;
        }
        __syncthreads();
        const float inv = sinv;
        const int r = k + 1 + tid;
        float lrk = 0.0f;
        if (r < N) {
            lrk = Zc[(size_t)r * N + k] * inv;
            Zc[(size_t)r * N + k] = lrk;
        }
        __threadfence_block();
        __syncthreads();
        if (r < N) {
            for (int c = k + 1; c <= r; ++c)
                Zc[(size_t)r * N + c] -= lrk * Zc[(size_t)c * N + k];
        }
        __threadfence_block();
        __syncthreads();
    }
    if (tid == 0) ldj_out[0] = lacc;
    // zero upper triangle so dense WMMA tiles see exact L
    for (int idx = tid; idx < N * N; idx += 1024) {
        const int rr = idx >> 10, cc = idx & 1023;
        if (cc > rr) Zc[idx] = 0.0f;
    }
}

// ---------------------------------------------------------------------------
// Kernel 4: sldj_out = sldj_in + ldj  (broadcast scalar)
// ---------------------------------------------------------------------------
__global__ void add_ldj(const float* __restrict__ sin_,
                        const float* __restrict__ ldj,
                        float* __restrict__ sout)
{
    const int idx = blockIdx.x * blockDim.x + threadIdx.x;
    if (idx < KM) sout[idx] = sin_[idx] + ldj[0];
}

// ---------------------------------------------------------------------------
// Kernel 5: C = Z_flat @ L^T via V_WMMA_F32_16X16X4_F32.
// 4x1 register blocking: each wave owns 4 row-tiles (64 batch rows) x 1 col
// tile; the B fragment (L row) is shared by 4 WMMAs per K-step. Triangular
// K-skip (L is lower-triangular with zeroed upper part). Scalar (wave-uniform)
// loop via readfirstlane so EXEC stays all-ones around WMMA. Output
// permutation [i -> (i%8)*128 + i/8] folded into column-tile choice so the
// 32 C stores are coalesced along t and clause-able with immediate offsets.
// ---------------------------------------------------------------------------
__global__ __launch_bounds__(256) void gemm_wmma(const float* __restrict__ Zf, // [16384,1024]
                                                 const float* __restrict__ L,  // [1024,1024] lower
                                                 float* __restrict__ Out)      // [16384][8][128]
{
    const int lane = threadIdx.x & 31;
    const int wave = threadIdx.x >> 5;
    const int w = blockIdx.x * 8 + wave;      // 0..16383 wave-tiles
    const int col = w & 63;                   // 64 column tiles
    const int rg  = w >> 6;                   // 256 row groups (64 rows each)
    const int d   = col & 7;
    const int t0  = (col >> 3) << 4;          // 0,16,...,112
    const size_t n0 = (size_t)rg << 6;        // 64 rows per group

    // B fragment: lane holds L row i = d + 8*(t0 + lane%16); K halves split at lane 16
    const int iL   = d + 8 * (t0 + (lane & 15));
    const int maxi = d + 8 * (t0 + 15);
    const int ktiles = __builtin_amdgcn_readfirstlane((maxi >> 2) + 1); // scalar loop
    const int khalf = (lane < 16) ? 0 : 2;

    const float* brow = L  + (size_t)iL * N_DT + khalf;
    const float* arow = Zf + (n0 + (lane & 15)) * N_DT + khalf;  // row-tile 0 base

    v8f c0 = {0.f,0.f,0.f,0.f,0.f,0.f,0.f,0.f};
    v8f c1 = c0, c2 = c0, c3 = c0;

#pragma unroll 2
    for (int kt = 0; kt < ktiles; ++kt) {
        const int o = kt * 4;
        v2f b  = *(const v2f*)(brow + o);
        v2f x0 = *(const v2f*)(arow + o);
        v2f x1 = *(const v2f*)(arow + o + 16 * N_DT);
        v2f x2 = *(const v2f*)(arow + o + 32 * N_DT);
        v2f x3 = *(const v2f*)(arow + o + 48 * N_DT);
        c0 = __builtin_amdgcn_wmma_f32_16x16x4_f32(false, x0, false, b, (short)0, c0, false, false);
        c1 = __builtin_amdgcn_wmma_f32_16x16x4_f32(false, x1, false, b, (short)0, c1, false, false);
        c2 = __builtin_amdgcn_wmma_f32_16x16x4_f32(false, x2, false, b, (short)0, c2, false, false);
        c3 = __builtin_amdgcn_wmma_f32_16x16x4_f32(false, x3, false, b, (short)0, c3, false, false);
    }

    // Store: lane column -> t = t0 + lane%16 (contiguous across lanes 0-15);
    // rows m = vgpr (+8 for upper half-lanes); 4 row tiles at +16-row strides.
    const int t = t0 + (lane & 15);
    const size_t rbase = n0 + ((lane < 16) ? 0 : 8);
    float* o = Out + rbase * N_DT + d * T_LEN + t;
#pragma unroll
    for (int v = 0; v < 8; ++v) o[(size_t)(v)      * N_DT] = c0[v];
#pragma unroll
    for (int v = 0; v < 8; ++v) o[(size_t)(v + 16) * N_DT] = c1[v];
#pragma unroll
    for (int v = 0; v < 8; ++v) o[(size_t)(v + 32) * N_DT] = c2[v];
#pragma unroll
    for (int v = 0; v < 8; ++v) o[(size_t)(v + 48) * N_DT] = c3[v];
}

// ---------------------------------------------------------------------------
extern "C" void kernel_launch(void* const* d_in, const int* in_sizes, int n_in,
                              void* d_out, int out_size, void* d_ws, size_t ws_size,
                              hipStream_t stream)
{
    const float* z         = (const float*)d_in[0];   // [16,1024,8,128]
    const float* sldj_in   = (const float*)d_in[1];   // [16,1024]
    const float* A         = (const float*)d_in[2];   // [2,8,8]
    const float* log_sigma = (const float*)d_in[3];   // [8]

    float* out_z    = (float*)d_out;                  // 16,777,216 floats
    float* out_sldj = out_z + (size_t)KM * N_DT;      // 16,384 floats

    float* ws    = (float*)d_ws;
    float* Zcov  = ws;                                // 1024*1024
    float* gamma = ws + (size_t)N_DT * N_DT;          // 128*64
    float* ldj   = gamma + T_LEN * 64;                // 1

    hipLaunchKernelGGL(init_cov, dim3(1), dim3(64), 0, stream, A, log_sigma, Zcov, gamma);
    for (int t = 1; t < T_LEN; ++t)
        hipLaunchKernelGGL(cov_row, dim3(T_LEN - t), dim3(64), 0, stream,
                           t, A, log_sigma, Zcov, gamma);
    hipLaunchKernelGGL(cholesky, dim3(1), dim3(1024), 0, stream, Zcov, ldj);
    hipLaunchKernelGGL(add_ldj, dim3(KM / 256), dim3(256), 0, stream, sldj_in, ldj, out_sldj);
    hipLaunchKernelGGL(gemm_wmma, dim3(16384 / 8), dim3(256), 0, stream, z, Zcov, out_z);
}